// SJLTProjection_44263932953119
// MI455X (gfx1250) — compile-verified
//
#include <hip/hip_runtime.h>
#include <hip/hip_bf16.h>

// SJLT scatter sketch, tuned for MI455X (gfx1250):
//  - memory-bound (192 MB streamed, ~8 us floor @ 23.3 TB/s)
//  - TDM (tensor_load_to_lds) double-buffered Global->LDS staging, TENSORcnt sync
//  - LDS f32 atomic scatter into bank-padded per-workgroup accumulator
//  - global_atomic_add_f32 flush into the 512 KB output (L2-resident)

#define TILE      512                 // features per staged tile (1 per thread)
#define NTHREADS  512                 // 16 wave32
#define BROWS     8                   // batch rows per workgroup
#define NGROUPS   4                   // 4 * 8 = 32 batch rows
#define PROJ      4096
#define C         8
#define ACC_STRIDE (PROJ + 8)         // pad: rows land in distinct LDS banks
#define CHUNK     (16 * TILE)         // 8192 features per workgroup

#define IDX_OFF   0                   // word offsets inside a staging buffer
#define SGN_OFF   (TILE * C)          // 4096 words
#define X_OFF     (2 * TILE * C)      // 8192 words
#define STAGE_WORDS (3 * TILE * C)    // 12288 words = 48 KB per buffer

#if defined(__HIP_DEVICE_COMPILE__) && __has_builtin(__builtin_amdgcn_tensor_load_to_lds)
#define USE_TDM 1
#else
#define USE_TDM 0
#endif

typedef unsigned int v4u __attribute__((ext_vector_type(4)));
typedef int          v4i __attribute__((ext_vector_type(4)));
typedef int          v8i __attribute__((ext_vector_type(8)));

#if USE_TDM
// Pack a 2D-tile Tensor DMA Descriptor (D#) and issue TENSOR_LOAD_TO_LDS.
// group0: count=1 | lds_addr | global_addr[56:0] | type=2
// group1: data_size=4B; tensor_dim0/1; tile_dim0/1; tensor_dim0_stride
__device__ __forceinline__ void tdm_issue_2d(unsigned ldsByteOff, const void* gptr,
                                             unsigned tileW, unsigned tileH,
                                             unsigned long long rowStrideElems) {
  unsigned long long ga = (unsigned long long)(uintptr_t)gptr;
  v4u g0;
  g0.x = 1u;                                                   // count=1, user desc
  g0.y = ldsByteOff;                                           // LDS byte address
  g0.z = (unsigned)ga;                                         // global addr [31:0]
  g0.w = (unsigned)((ga >> 32) & 0x01FFFFFFu) | (2u << 30);    // addr[56:32] | type=2
  v8i g1;
  g1[0] = (int)(2u << 16);                                     // wg_mask=0, data_size=4B
  g1[1] = (int)((tileW & 0xFFFFu) << 16);                      // tensor_dim0[15:0]
  g1[2] = (int)((tileW >> 16) | ((tileH & 0xFFFFu) << 16));    // td0[31:16] | td1[15:0]
  g1[3] = (int)((tileH >> 16) | (tileW << 16));                // td1[31:16] | tile_dim0
  g1[4] = (int)tileH;                                          // tile_dim1 | tile_dim2=0
  g1[5] = (int)(unsigned)(rowStrideElems & 0xFFFFFFFFull);     // dim0_stride[31:0]
  g1[6] = (int)(unsigned)((rowStrideElems >> 32) & 0xFFFFull); // dim0_stride[47:32]
  g1[7] = 0;                                                   // dim1_stride unused (2D)
  v4i z4 = {0, 0, 0, 0};
#if defined(__clang_major__) && (__clang_major__ >= 23)
  v8i z8 = {0, 0, 0, 0, 0, 0, 0, 0};
  __builtin_amdgcn_tensor_load_to_lds(g0, g1, z4, z4, z8, 0);
#else
  __builtin_amdgcn_tensor_load_to_lds(g0, g1, z4, z4, 0);
#endif
}
#endif

// ---------------------------------------------------------------------------
// TDM probe: never launched. Defined FIRST so the disasm snippet (which shows
// the start of the module) displays the tensor_load_to_lds / s_wait_tensorcnt
// lowering for visual verification.
// ---------------------------------------------------------------------------
__global__ void tdm_probe_kernel(const float* __restrict__ g, float* __restrict__ o) {
  __shared__ __attribute__((aligned(16))) float buf[256];
#if USE_TDM
  tdm_issue_2d((unsigned)(uintptr_t)&buf[0], g, 256u, 1u, 256ull);
  __builtin_amdgcn_s_wait_tensorcnt(0);
#else
  buf[threadIdx.x] = g[threadIdx.x];
#endif
  __syncthreads();
  o[threadIdx.x] = buf[threadIdx.x];
}

__global__ __launch_bounds__(NTHREADS)
void sjlt_scatter_kernel(const float* __restrict__ x,
                         const int* __restrict__ idxg,
                         const int* __restrict__ sgng,
                         float* __restrict__ out, int D) {
  __shared__ __attribute__((aligned(16))) float    acc[BROWS * ACC_STRIDE]; // ~128 KB
  __shared__ __attribute__((aligned(16))) unsigned stage[2][STAGE_WORDS];   // 2 x 48 KB

  const int tid = threadIdx.x;
  const int bg  = blockIdx.y;                 // batch group: rows bg*8 .. bg*8+7
  const int cb  = blockIdx.x * CHUNK;
  const int ce  = min(D, cb + CHUNK);
  const int ntiles = (ce - cb + TILE - 1) / TILE;

  // zero the private accumulator
  for (int i = tid; i < BROWS * ACC_STRIDE; i += NTHREADS) acc[i] = 0.0f;

#if USE_TDM
  auto issue_tile = [&](int buf, int d0, int w) {
    unsigned base = (unsigned)(uintptr_t)&stage[buf][0];   // flat->LDS offset (low 32b)
    tdm_issue_2d(base + IDX_OFF * 4u, idxg + (size_t)d0 * C,
                 (unsigned)(w * C), 1u, (unsigned long long)(w * C));
    tdm_issue_2d(base + SGN_OFF * 4u, sgng + (size_t)d0 * C,
                 (unsigned)(w * C), 1u, (unsigned long long)(w * C));
    tdm_issue_2d(base + X_OFF * 4u, x + (size_t)(bg * BROWS) * D + d0,
                 (unsigned)w, (unsigned)BROWS, (unsigned long long)D);
  };
  if (tid < 32) {                             // wave 0 only: TDM is per-wave
    issue_tile(0, cb, min(TILE, ce - cb));    // prologue: tile 0 -> buf 0
  }
#endif

  int cur = 0;
  for (int it = 0; it < ntiles; ++it) {
    const int d0 = cb + it * TILE;
    const int w  = min(TILE, ce - d0);

#if USE_TDM
    if (tid < 32) {
      if (it + 1 < ntiles) {
        const int d1 = d0 + TILE;
        issue_tile(cur ^ 1, d1, min(TILE, ce - d1));   // prefetch next tile
        __builtin_amdgcn_s_wait_tensorcnt(3);          // current tile's 3 DMAs done
      } else {
        __builtin_amdgcn_s_wait_tensorcnt(0);
      }
    }
#else
    { // fallback: cooperative synchronous staging (same LDS layout)
      for (int i = tid; i < w * C; i += NTHREADS) {
        stage[cur][IDX_OFF + i] = (unsigned)idxg[(size_t)d0 * C + i];
        stage[cur][SGN_OFF + i] = (unsigned)sgng[(size_t)d0 * C + i];
      }
      for (int i = tid; i < BROWS * w; i += NTHREADS) {
        int b = i / w, e = i - b * w;
        stage[cur][X_OFF + i] =
            __float_as_uint(x[(size_t)(bg * BROWS + b) * D + d0 + e]);
      }
    }
#endif
    __syncthreads();   // buf[cur] visible to all waves (and acc zeroed on iter 0)

    if (tid < w) {
      const unsigned* tI = &stage[cur][IDX_OFF];
      const unsigned* tS = &stage[cur][SGN_OFF];
      const unsigned* tX = &stage[cur][X_OFF];
      uint4 i0 = *(const uint4*)(tI + tid * C);
      uint4 i1 = *(const uint4*)(tI + tid * C + 4);
      uint4 s0 = *(const uint4*)(tS + tid * C);
      uint4 s1 = *(const uint4*)(tS + tid * C + 4);
      const unsigned ii[C] = {i0.x, i0.y, i0.z, i0.w, i1.x, i1.y, i1.z, i1.w};
      const unsigned sm[C] = {s0.x & 0x80000000u, s0.y & 0x80000000u,
                              s0.z & 0x80000000u, s0.w & 0x80000000u,
                              s1.x & 0x80000000u, s1.y & 0x80000000u,
                              s1.z & 0x80000000u, s1.w & 0x80000000u};
#pragma unroll
      for (int b = 0; b < BROWS; ++b) {
        const unsigned xb = tX[b * w + tid];
        float* arow = &acc[b * ACC_STRIDE];
#pragma unroll
        for (int j = 0; j < C; ++j) {
          atomicAdd(&arow[ii[j]], __uint_as_float(xb ^ sm[j]));  // ds_add_f32
        }
      }
    }
    __syncthreads();   // buf[cur] fully consumed before DMA overwrites it
    cur ^= 1;
  }

  // flush: scaled partial rows into global output (L2-resident atomics)
  const float scale = 0.35355339059327373f;  // 1/sqrt(8)
  for (int i = tid; i < BROWS * PROJ; i += NTHREADS) {
    const int b = i >> 12;            // / 4096
    const int p = i & (PROJ - 1);
    atomicAdd(&out[(size_t)(bg * BROWS + b) * PROJ + p],
              acc[b * ACC_STRIDE + p] * scale);
  }
}

extern "C" void kernel_launch(void* const* d_in, const int* in_sizes, int n_in,
                              void* d_out, int out_size, void* d_ws, size_t ws_size,
                              hipStream_t stream) {
  const float* x   = (const float*)d_in[0];
  const int*   idx = (const int*)d_in[1];
  const int*   sgn = (const int*)d_in[2];
  const int D = in_sizes[1] / C;                 // 1,000,000

  hipMemsetAsync(d_out, 0, (size_t)out_size * sizeof(float), stream);

  dim3 grid((unsigned)((D + CHUNK - 1) / CHUNK), NGROUPS, 1);
  sjlt_scatter_kernel<<<grid, NTHREADS, 0, stream>>>(x, idx, sgn, (float*)d_out, D);
}